// Canny_55439437857048
// MI455X (gfx1250) — compile-verified
//
#include <hip/hip_runtime.h>
#include <math.h>

typedef __attribute__((ext_vector_type(2))) float v2f;
typedef __attribute__((ext_vector_type(8))) float v8f;

#define IMG_H 1024
#define IMG_W 1024
#define IMG_HW (IMG_H * IMG_W)

// jnp.pad 'reflect' (mirror excluding edge) for pad <= 2
__device__ __forceinline__ int reflect2(int p, int n) {
    p = (p < 0) ? -p : p;
    p = (p >= n) ? (2 * n - 2 - p) : p;
    return p;
}

// normalized gaussian(sigma=1, ksize=5) tap, 0 outside [0,4]; branch-free selects
__device__ __forceinline__ float gcoef(int d) {
    float r = 0.0f;
    r = (d == 0 || d == 4) ? 0.054488685f : r;
    r = (d == 1 || d == 3) ? 0.244201342f : r;
    r = (d == 2)           ? 0.402619947f : r;
    return r;
}

// ---------------- grayscale ----------------
__global__ void k_gray(const float* __restrict__ x, float* __restrict__ g, int np) {
    int i = blockIdx.x * blockDim.x + threadIdx.x;
    if (i >= np) return;
    int b = i >> 20;            // / IMG_HW (2^20)
    int p = i & (IMG_HW - 1);
    const float* base = x + (size_t)b * 3 * IMG_HW + p;
    g[i] = 0.299f * base[0] + 0.587f * base[IMG_HW] + 0.114f * base[2 * IMG_HW];
}

// ---------------- horizontal gaussian via WMMA f32 16x16x4 ----------------
// Out(16x16) = A(16x20) * S(20x16), S[k][n] = g[k-n]; 5 accumulating K=4 steps.
__global__ void k_blur_x(const float* __restrict__ in, float* __restrict__ out) {
    int wave = (int)((blockIdx.x * blockDim.x + threadIdx.x) >> 5);
    int lane = threadIdx.x & 31;
    int half = lane >> 4;       // which K-pair this lane holds
    int idx  = lane & 15;       // M (for A) / N (for B,D)
    int b = wave >> 12;         // 4096 tiles per 1024x1024 plane
    int t = wave & 4095;
    int row0 = (t >> 6) << 4;
    int col0 = (t & 63) << 4;
    const float* img = in + (size_t)b * IMG_HW;
    float* dst = out + (size_t)b * IMG_HW;
    v8f acc = {};
    int arow = row0 + idx;
#pragma unroll
    for (int s = 0; s < 5; ++s) {
        int k0 = 4 * s + 2 * half;   // padded-column index this lane's A pair covers
        v2f a, bb;
        a.x = img[arow * IMG_W + reflect2(col0 - 2 + k0, IMG_W)];
        a.y = img[arow * IMG_W + reflect2(col0 - 1 + k0, IMG_W)];
        bb.x = gcoef(k0 - idx);       // S[k0  ][idx]
        bb.y = gcoef(k0 + 1 - idx);   // S[k0+1][idx]
        acc = __builtin_amdgcn_wmma_f32_16x16x4_f32(
            false, a, false, bb, (short)0, acc, false, false);
    }
#pragma unroll
    for (int j = 0; j < 8; ++j)       // D: VGPR j -> M = j + 8*half
        dst[(row0 + 8 * half + j) * IMG_W + col0 + idx] = acc[j];
}

// ---------------- vertical gaussian via WMMA f32 16x16x4 ----------------
// Out(16x16) = T(16x20) * B(20x16), T[m][k] = g[k-m], B = padded input rows.
__global__ void k_blur_y(const float* __restrict__ in, float* __restrict__ out) {
    int wave = (int)((blockIdx.x * blockDim.x + threadIdx.x) >> 5);
    int lane = threadIdx.x & 31;
    int half = lane >> 4;
    int idx  = lane & 15;
    int b = wave >> 12;
    int t = wave & 4095;
    int row0 = (t >> 6) << 4;
    int col0 = (t & 63) << 4;
    const float* img = in + (size_t)b * IMG_HW;
    float* dst = out + (size_t)b * IMG_HW;
    v8f acc = {};
#pragma unroll
    for (int s = 0; s < 5; ++s) {
        int k0 = 4 * s + 2 * half;
        v2f a, bb;
        a.x = gcoef(k0 - idx);        // T[idx][k0]
        a.y = gcoef(k0 + 1 - idx);    // T[idx][k0+1]
        bb.x = img[reflect2(row0 - 2 + k0, IMG_H) * IMG_W + col0 + idx];
        bb.y = img[reflect2(row0 - 1 + k0, IMG_H) * IMG_W + col0 + idx];
        acc = __builtin_amdgcn_wmma_f32_16x16x4_f32(
            false, a, false, bb, (short)0, acc, false, false);
    }
#pragma unroll
    for (int j = 0; j < 8; ++j)
        dst[(row0 + 8 * half + j) * IMG_W + col0 + idx] = acc[j];
}

// ---------------- sobel + magnitude + quantized direction ----------------
__global__ void k_sobel(const float* __restrict__ blur, float* __restrict__ mag,
                        unsigned char* __restrict__ dir, int np) {
    int i = blockIdx.x * blockDim.x + threadIdx.x;
    if (i >= np) return;
    int b = i >> 20, p = i & (IMG_HW - 1);
    int y = p >> 10, x = p & (IMG_W - 1);
    const float* im = blur + (size_t)b * IMG_HW;
    int ym = (y > 0) ? y - 1 : 0, yp = (y < IMG_H - 1) ? y + 1 : IMG_H - 1;
    int xm = (x > 0) ? x - 1 : 0, xp = (x < IMG_W - 1) ? x + 1 : IMG_W - 1;
    float a00 = im[ym * IMG_W + xm], a01 = im[ym * IMG_W + x], a02 = im[ym * IMG_W + xp];
    float a10 = im[y  * IMG_W + xm],                           a12 = im[y  * IMG_W + xp];
    float a20 = im[yp * IMG_W + xm], a21 = im[yp * IMG_W + x], a22 = im[yp * IMG_W + xp];
    float gx = (a02 - a00) + 2.f * (a12 - a10) + (a22 - a20);
    float gy = (a20 - a00) + 2.f * (a21 - a01) + (a22 - a02);
    float m = sqrtf(gx * gx + gy * gy + 1e-6f);
    int q = (int)rintf(atan2f(gy, gx) * 1.27323954474f);  // *(180/pi)/45, q in [-4,4]
    mag[i] = m;
    dir[i] = (unsigned char)((q + 8) & 7);                 // mod 8
}

// neighbor offset in kornia order: 0=right, clockwise
__device__ __forceinline__ float nms_nbr(const float* __restrict__ mg, int y, int x, int d) {
    int dr = (d >= 1 && d <= 3) ? 1 : ((d >= 5) ? -1 : 0);
    int dc = (d == 0 || d == 1 || d == 7) ? 1 : ((d >= 3 && d <= 5) ? -1 : 0);
    int ny = y + dr, nx = x + dc;
    bool ok = (ny >= 0) && (ny < IMG_H) && (nx >= 0) && (nx < IMG_W);
    return ok ? mg[ny * IMG_W + nx] : 0.0f;   // zero padding in reference conv
}

// ---------------- NMS + double threshold ----------------
__global__ void k_nms(const float* __restrict__ mag, const unsigned char* __restrict__ dir,
                      float* __restrict__ out_mag, unsigned char* __restrict__ e, int np) {
    int i = blockIdx.x * blockDim.x + threadIdx.x;
    if (i >= np) return;
    int b = i >> 20, p = i & (IMG_HW - 1);
    int y = p >> 10, x = p & (IMG_W - 1);
    const float* mg = mag + (size_t)b * IMG_HW;
    float m = mg[p];
    int pos = dir[i];
    int neg = pos ^ 4;
    // is_max = min(m - nbr(pos), m - nbr(neg)) > 0
    bool is_max = ((m - nms_nbr(mg, y, x, pos)) > 0.0f) &&
                  ((m - nms_nbr(mg, y, x, neg)) > 0.0f);
    float mm = is_max ? m : 0.0f;
    out_mag[i] = mm;
    e[i] = (mm > 0.2f) ? (unsigned char)2 : ((mm > 0.1f) ? (unsigned char)1 : (unsigned char)0);
}

// ---------------- LDS-tiled hysteresis: intra-tile flood fill to convergence ----
// 64x64 tile + 1px halo in LDS; monotone 1->2 upgrades (Gauss-Seidel, race-safe),
// block-wide early-exit flag. Cross-tile propagation via outer ping-pong passes.
#define TILE 64
#define LDS_H (TILE + 2)
#define LDS_W (TILE + 2)
#define LDS_STRIDE (TILE + 4)   // 68: break 64B alignment between rows

__global__ void k_hyst_tile(const unsigned char* __restrict__ src,
                            unsigned char* __restrict__ dst) {
    __shared__ unsigned char S[LDS_H * LDS_STRIDE];
    __shared__ int changed;
    int tid = threadIdx.x;
    int tile = blockIdx.x;
    int b = tile >> 8;                 // 256 tiles per 1024x1024 plane
    int t = tile & 255;
    int ty0 = (t >> 4) << 6;
    int tx0 = (t & 15) << 6;
    const unsigned char* sp = src + (size_t)b * IMG_HW;
    unsigned char* dp = dst + (size_t)b * IMG_HW;

    // cooperative load of (TILE+2)^2 halo region; OOB -> 0 (zero padding)
    for (int l = tid; l < LDS_H * LDS_W; l += 256) {
        int ly = l / LDS_W, lx = l - ly * LDS_W;
        int gy = ty0 + ly - 1, gx = tx0 + lx - 1;
        bool ok = (gy >= 0) && (gy < IMG_H) && (gx >= 0) && (gx < IMG_W);
        S[ly * LDS_STRIDE + lx] = ok ? sp[gy * IMG_W + gx] : (unsigned char)0;
    }
    __syncthreads();

    int r = tid >> 2;                  // owned row 0..63
    int cbase = (tid & 3) << 4;        // owned 16-px row segment
    int ly = r + 1;

    for (int it = 0; it < 64; ++it) {
        if (tid == 0) changed = 0;
        __syncthreads();
        int local = 0;
#pragma unroll
        for (int j = 0; j < 16; ++j) {
            int lx = cbase + 1 + j;
            unsigned char v = S[ly * LDS_STRIDE + lx];
            if (v == 1) {
                const unsigned char* up = &S[(ly - 1) * LDS_STRIDE + lx];
                const unsigned char* md = &S[ ly      * LDS_STRIDE + lx];
                const unsigned char* dn = &S[(ly + 1) * LDS_STRIDE + lx];
                bool any = (up[-1] == 2) || (up[0] == 2) || (up[1] == 2) ||
                           (md[-1] == 2) ||                 (md[1] == 2) ||
                           (dn[-1] == 2) || (dn[0] == 2) || (dn[1] == 2);
                if (any) { S[ly * LDS_STRIDE + lx] = 2; local = 1; }
            }
        }
        if (local) changed = 1;        // benign same-value race
        __syncthreads();
        int c = changed;
        __syncthreads();               // all reads of flag done before next reset
        if (!c) break;
    }

    // write interior back
#pragma unroll
    for (int j = 0; j < 16; ++j)
        dp[(ty0 + r) * IMG_W + tx0 + cbase + j] = S[ly * LDS_STRIDE + cbase + 1 + j];
}

// ---------------- final edges = (state == strong) ----------------
__global__ void k_final(const unsigned char* __restrict__ e, float* __restrict__ out, int np) {
    int i = blockIdx.x * blockDim.x + threadIdx.x;
    if (i >= np) return;
    out[i] = (e[i] == 2) ? 1.0f : 0.0f;
}

extern "C" void kernel_launch(void* const* d_in, const int* in_sizes, int n_in,
                              void* d_out, int out_size, void* d_ws, size_t ws_size,
                              hipStream_t stream) {
    (void)n_in; (void)out_size; (void)ws_size;
    const float* x = (const float*)d_in[0];
    int B = in_sizes[0] / (3 * IMG_HW);   // 4
    int np = B * IMG_HW;                  // 4M pixels

    // workspace carve-up (all fully overwritten before reads)
    char* ws = (char*)d_ws;
    size_t fbytes = (size_t)np * sizeof(float);
    auto up = [](size_t v) { return (v + 255) & ~(size_t)255; };
    size_t o = 0;
    float* buf0 = (float*)(ws + o);            o += up(fbytes);  // gray, then blurred
    float* buf1 = (float*)(ws + o);            o += up(fbytes);  // blur-x tmp, then magnitude
    unsigned char* dir = (unsigned char*)(ws + o); o += up((size_t)np);
    unsigned char* eA  = (unsigned char*)(ws + o); o += up((size_t)np);
    unsigned char* eB  = (unsigned char*)(ws + o);

    float* out_mag   = (float*)d_out;
    float* out_edges = (float*)d_out + np;

    const int tpb = 256;
    int blocksPix = (np + tpb - 1) / tpb;

    k_gray<<<blocksPix, tpb, 0, stream>>>(x, buf0, np);

    // one 16x16 tile per wave, 8 waves per block; exact cover -> EXEC all ones
    int waveTiles = B * (IMG_H / 16) * (IMG_W / 16);
    int blocksTile = waveTiles / 8;
    k_blur_x<<<blocksTile, tpb, 0, stream>>>(buf0, buf1);
    k_blur_y<<<blocksTile, tpb, 0, stream>>>(buf1, buf0);

    k_sobel<<<blocksPix, tpb, 0, stream>>>(buf0, buf1, dir, np);
    k_nms<<<blocksPix, tpb, 0, stream>>>(buf1, dir, out_mag, eA, np);

    // outer ping-pong passes; each pass converges intra-tile in LDS.
    // 8 passes -> cross-tile propagation of ~8 tiles (512 px chain length).
    int blocksTileH = B * (IMG_H / TILE) * (IMG_W / TILE);   // 1024
    const int OUTER = 8;  // even -> final state lands in eA
    for (int it = 0; it < OUTER; ++it) {
        const unsigned char* s = (it & 1) ? eB : eA;
        unsigned char* d       = (it & 1) ? eA : eB;
        k_hyst_tile<<<blocksTileH, tpb, 0, stream>>>(s, d);
    }
    k_final<<<blocksPix, tpb, 0, stream>>>(eA, out_edges, np);
}